// DiversificationBlock_30468497998623
// MI455X (gfx1250) — compile-verified
//
#include <hip/hip_runtime.h>

// DiversificationBlock for MI455X (gfx1250).
// Memory-bound elementwise op + per-(b,c) spatial max.
// Min traffic: act 102.8MB + rc 102.8MB + pmask 10.6MB + out 102.8MB ~= 319MB
// -> ~13.7us at 23.3 TB/s. No matmul -> no WMMA; the CDNA5 feature that pays
// here is async global->LDS staging (read activation from HBM exactly once
// even though the algorithm needs two passes over it).

#define HW     3136   // 56*56 spatial elements per (b,c)
#define W      56
#define COV    54     // 18*3 covered region
#define PM_N   324    // 18*18 patch mask elements per (b,c)
#define PM_W   18
#define NT     256    // 8 wave32 waves
#define ALPHA  0.1f

// Types matching the async-to-LDS builtin signatures (from hipcc diagnostics):
//   b128: (v4i AS1*, v4i AS3*, imm, imm)   b32: (int AS1*, int AS3*, imm, imm)
typedef int v4i_t __attribute__((vector_size(16)));
typedef __attribute__((address_space(1))) v4i_t* glb_v4i_p;
typedef __attribute__((address_space(3))) v4i_t* lds_v4i_p;
typedef __attribute__((address_space(1))) int*   glb_i32_p;
typedef __attribute__((address_space(3))) int*   lds_i32_p;

__device__ __forceinline__ float dv_elem(float a, int r, int pmv, float peak) {
    // bc = rc*pc + patch*(1-pc); apply alpha iff bc >= 1
    const bool pc   = (a == peak);
    const bool cond = pc ? (r >= 1) : (pmv >= 1);
    return cond ? a * ALPHA : a;
}

__global__ __launch_bounds__(NT)
void diversification_kernel(const float* __restrict__ act,
                            const int*   __restrict__ rc,
                            const int*   __restrict__ pmask,
                            float*       __restrict__ out) {
    __shared__ __align__(16) float act_s[HW];   // 12544 B
    __shared__ int   pm_s[PM_N];                //  1296 B
    __shared__ float red_s[NT / 32];

    const int       tid = threadIdx.x;
    const long long bc  = blockIdx.x;

    float* gact = const_cast<float*>(act)   + bc * (long long)HW;
    int*   gpm  = const_cast<int*>(pmask)   + bc * (long long)PM_N;

    // ---- Phase 1: async DMA the tile into LDS (ASYNCcnt-tracked) ----
    for (int i = tid; i < HW / 4; i += NT) {
        __builtin_amdgcn_global_load_async_to_lds_b128(
            (glb_v4i_p)(gact + 4 * i), (lds_v4i_p)(act_s + 4 * i), 0, 0);
    }
    for (int i = tid; i < PM_N; i += NT) {
        __builtin_amdgcn_global_load_async_to_lds_b32(
            (glb_i32_p)(gpm + i), (lds_i32_p)(pm_s + i), 0, 0);
    }
    __builtin_amdgcn_s_wait_asynccnt(0);
    __syncthreads();

    // ---- Phase 2: per-(b,c) spatial max (wave32 butterfly + LDS combine) ----
    float m = -3.402823466e38f;
    for (int i = tid; i < HW; i += NT) m = fmaxf(m, act_s[i]);
#pragma unroll
    for (int off = 16; off > 0; off >>= 1)
        m = fmaxf(m, __shfl_xor(m, off, 32));
    if ((tid & 31) == 0) red_s[tid >> 5] = m;
    __syncthreads();
    if (tid == 0) {
        float p = red_s[0];
#pragma unroll
        for (int w = 1; w < NT / 32; ++w) p = fmaxf(p, red_s[w]);
        red_s[0] = p;
    }
    __syncthreads();
    const float peak = red_s[0];

    // ---- Phase 3: elementwise, float4-vectorized (b128 loads/stores) ----
    const int4*   grc  = (const int4*)(rc + bc * (long long)HW);
    float4*       gout = (float4*)(out + bc * (long long)HW);
    const float4* act4 = (const float4*)act_s;
    for (int i = tid; i < HW / 4; i += NT) {
        const float4 a = act4[i];
        const int4   r = grc[i];
        const int lin = i * 4;
        const int y   = lin / W;
        const int x   = lin - y * W;          // W % 4 == 0 -> same row for all 4
        const int pr  = (y < COV) ? (y / 3) * PM_W : -1;
        const int p0  = (pr >= 0 && x + 0 < COV) ? pm_s[pr + (x + 0) / 3] : 0;
        const int p1  = (pr >= 0 && x + 1 < COV) ? pm_s[pr + (x + 1) / 3] : 0;
        const int p2  = (pr >= 0 && x + 2 < COV) ? pm_s[pr + (x + 2) / 3] : 0;
        const int p3  = (pr >= 0 && x + 3 < COV) ? pm_s[pr + (x + 3) / 3] : 0;
        float4 o;
        o.x = dv_elem(a.x, r.x, p0, peak);
        o.y = dv_elem(a.y, r.y, p1, peak);
        o.z = dv_elem(a.z, r.z, p2, peak);
        o.w = dv_elem(a.w, r.w, p3, peak);
        gout[i] = o;
    }
}

extern "C" void kernel_launch(void* const* d_in, const int* in_sizes, int n_in,
                              void* d_out, int out_size, void* d_ws, size_t ws_size,
                              hipStream_t stream) {
    const float* act   = (const float*)d_in[0];  // [32,256,56,56] f32
    const int*   rc    = (const int*)d_in[1];    // [32,256,56,56] i32
    const int*   pmask = (const int*)d_in[2];    // [32,256,18,18] i32
    float*       out   = (float*)d_out;

    const int nbc = in_sizes[0] / HW;            // 32*256 = 8192 (b,c) slices
    diversification_kernel<<<nbc, NT, 0, stream>>>(act, rc, pmask, out);
}